// GATLayer_67439576482327
// MI455X (gfx1250) — compile-verified
//
#include <hip/hip_runtime.h>
#include <math.h>

#define NODE_DIM 128
#define OUT_DIM 64
#define NEG_SLOPE 0.01f

typedef __attribute__((ext_vector_type(2))) float v2f;
typedef __attribute__((ext_vector_type(8))) float v8f;

// ---------------------------------------------------------------------------
// Init: zero the output accumulator, set segment-max to -inf, denom to 0.
// ---------------------------------------------------------------------------
__global__ void gat_init(float* __restrict__ out, float* __restrict__ m,
                         float* __restrict__ denom, int n_nodes, int out_elems) {
    int i = blockIdx.x * blockDim.x + threadIdx.x;
    if (i < out_elems) out[i] = 0.0f;
    if (i < n_nodes) { m[i] = -INFINITY; denom[i] = 0.0f; }
}

// ---------------------------------------------------------------------------
// z = h @ W_fc using V_WMMA_F32_16X16X4_F32. One wave32 per 16x16 tile of z.
// N % 16 == 0 and OUT_DIM == 64, so no tails: EXEC is all-ones in WMMA waves.
//
// A 16x4 layout: lanes 0-15 -> M=lane, K={0,1}; lanes 16-31 -> M=lane-16, K={2,3}
// B 4x16 layout: VGPR j holds K-row (2*half + j), lanes select N
// D layout: VGPR r -> row r (lanes 0-15) / row 8+r (lanes 16-31), col = lane%16
// ---------------------------------------------------------------------------
__global__ void gat_gemm_wmma(const float* __restrict__ h,
                              const float* __restrict__ W,
                              float* __restrict__ z, int mtiles) {
    const int wid  = (blockIdx.x * blockDim.x + threadIdx.x) >> 5;
    const int lane = threadIdx.x & 31;
    const int NT   = OUT_DIM / 16;                 // 4 column tiles
    if (wid >= mtiles * NT) return;                // whole-wave uniform exit
    const int mt   = wid / NT;
    const int nt   = wid - mt * NT;
    const int half = lane >> 4;                    // 0 or 1
    const int l16  = lane & 15;

    const float* arow = h + (size_t)(mt * 16 + l16) * NODE_DIM + 2 * half;
    const float* bptr = W + (size_t)(2 * half) * OUT_DIM + nt * 16 + l16;

    v8f acc = {};
#pragma unroll
    for (int k = 0; k < NODE_DIM; k += 4) {
        v2f a = *(const v2f*)(arow + k);           // K = k+2*half, k+2*half+1
        v2f b;
        b.x = bptr[(size_t)k * OUT_DIM];           // B[K=k+2*half  ][n]
        b.y = bptr[(size_t)k * OUT_DIM + OUT_DIM]; // B[K=k+2*half+1][n]
        acc = __builtin_amdgcn_wmma_f32_16x16x4_f32(
            /*neg_a=*/false, a, /*neg_b=*/false, b,
            /*c_mod=*/(short)0, acc, /*reuse_a=*/false, /*reuse_b=*/false);
    }

    float* zout = z + (size_t)(mt * 16 + 8 * half) * OUT_DIM + nt * 16 + l16;
#pragma unroll
    for (int r = 0; r < 8; ++r) zout[(size_t)r * OUT_DIM] = acc[r];
}

// ---------------------------------------------------------------------------
// s_src[n] = dot(z[n], Wa[0:64]);  s_dst[n] = dot(z[n], Wa[64:128])
// One wave32 per node: float2 per lane + shuffle tree reduction.
// ---------------------------------------------------------------------------
__global__ void gat_scores(const float* __restrict__ z,
                           const float* __restrict__ Wa,
                           float* __restrict__ s_src,
                           float* __restrict__ s_dst, int n_nodes) {
    int wid  = (blockIdx.x * blockDim.x + threadIdx.x) >> 5;
    int lane = threadIdx.x & 31;
    if (wid >= n_nodes) return;
    v2f zz = *(const v2f*)(z + (size_t)wid * OUT_DIM + 2 * lane);
    v2f wa = *(const v2f*)(Wa + 2 * lane);
    v2f wb = *(const v2f*)(Wa + OUT_DIM + 2 * lane);
    float a0 = zz.x * wa.x + zz.y * wa.y;
    float a1 = zz.x * wb.x + zz.y * wb.y;
#pragma unroll
    for (int off = 16; off > 0; off >>= 1) {
        a0 += __shfl_down(a0, off, 32);
        a1 += __shfl_down(a1, off, 32);
    }
    if (lane == 0) { s_src[wid] = a0; s_dst[wid] = a1; }
}

// Ordered-int float atomic max (correct for mixed signs; init must be -inf).
__device__ inline void atomicMaxFloat(float* addr, float value) {
    if (value >= 0.0f)
        atomicMax((int*)addr, __float_as_int(value));
    else
        atomicMin((unsigned int*)addr, __float_as_uint(value));
}

// ---------------------------------------------------------------------------
// e[i] = leaky_relu(s_src[src] + s_dst[dst]); segment max into m[dst].
// ---------------------------------------------------------------------------
__global__ void gat_edge_score(const int* __restrict__ src,
                               const int* __restrict__ dst,
                               const float* __restrict__ s_src,
                               const float* __restrict__ s_dst,
                               float* __restrict__ e,
                               float* __restrict__ m, int n_edges) {
    int i = blockIdx.x * blockDim.x + threadIdx.x;
    if (i >= n_edges) return;
    float v = s_src[src[i]] + s_dst[dst[i]];
    v = (v > 0.0f) ? v : NEG_SLOPE * v;
    e[i] = v;
    atomicMaxFloat(m + dst[i], v);
}

// ---------------------------------------------------------------------------
// e[i] <- exp(e[i] - m[dst]); denom[dst] += e[i]
// ---------------------------------------------------------------------------
__global__ void gat_edge_exp(const int* __restrict__ dst,
                             const float* __restrict__ m,
                             float* __restrict__ e,
                             float* __restrict__ denom, int n_edges) {
    int i = blockIdx.x * blockDim.x + threadIdx.x;
    if (i >= n_edges) return;
    float ex = expf(e[i] - m[dst[i]]);
    e[i] = ex;
    atomicAdd(denom + dst[i], ex);
}

// ---------------------------------------------------------------------------
// out[dst] += (e[i]/denom[dst]) * z[src].  One wave32 per edge; each lane
// owns 2 of the 64 output dims (coalesced float2 gather of z[src]).
// ---------------------------------------------------------------------------
__global__ void gat_scatter(const int* __restrict__ src,
                            const int* __restrict__ dst,
                            const float* __restrict__ e,
                            const float* __restrict__ denom,
                            const float* __restrict__ z,
                            float* __restrict__ out, int n_edges) {
    int wid  = (blockIdx.x * blockDim.x + threadIdx.x) >> 5;
    int lane = threadIdx.x & 31;
    if (wid >= n_edges) return;
    int s = src[wid], d = dst[wid];
    float alpha = e[wid] / denom[d];
    v2f zv = *(const v2f*)(z + (size_t)s * OUT_DIM + 2 * lane);
    float* o = out + (size_t)d * OUT_DIM + 2 * lane;
    atomicAdd(o,     alpha * zv.x);
    atomicAdd(o + 1, alpha * zv.y);
}

// ---------------------------------------------------------------------------
extern "C" void kernel_launch(void* const* d_in, const int* in_sizes, int n_in,
                              void* d_out, int out_size, void* d_ws, size_t ws_size,
                              hipStream_t stream) {
    const float* h    = (const float*)d_in[0];
    const int*   src  = (const int*)d_in[1];
    const int*   dst  = (const int*)d_in[2];
    const float* Wfc  = (const float*)d_in[3];
    const float* Watt = (const float*)d_in[4];
    float*       out  = (float*)d_out;

    const int N = in_sizes[0] / NODE_DIM;   // 50000
    const int E = in_sizes[1];              // 800000

    // Workspace carve-out (256B aligned slices)
    char*  ws  = (char*)d_ws;
    size_t off = 0;
    auto carve = [&](size_t bytes) -> float* {
        float* p = (float*)(ws + off);
        off = (off + bytes + 255) & ~(size_t)255;
        return p;
    };
    float* z     = carve((size_t)N * OUT_DIM * sizeof(float));
    float* s_src = carve((size_t)N * sizeof(float));
    float* s_dst = carve((size_t)N * sizeof(float));
    float* m     = carve((size_t)N * sizeof(float));
    float* denom = carve((size_t)N * sizeof(float));
    float* e     = carve((size_t)E * sizeof(float));
    (void)ws_size;

    const int BT = 256;

    // 1) init out / m / denom
    {
        int elems  = N * OUT_DIM;
        int blocks = (elems + BT - 1) / BT;
        gat_init<<<blocks, BT, 0, stream>>>(out, m, denom, N, elems);
    }
    // 2) z = h @ W_fc  (WMMA, one wave per 16x16 tile)
    {
        int mtiles = N / 16;                       // 3125 (exact)
        int waves  = mtiles * (OUT_DIM / 16);      // 12500
        int blocks = (waves * 32 + BT - 1) / BT;
        gat_gemm_wmma<<<blocks, BT, 0, stream>>>(h, Wfc, z, mtiles);
    }
    // 3) per-node attention scores (one wave per node)
    {
        int blocks = (N * 32 + BT - 1) / BT;
        gat_scores<<<blocks, BT, 0, stream>>>(z, Watt, s_src, s_dst, N);
    }
    // 4) edge scores + segment max
    {
        int blocks = (E + BT - 1) / BT;
        gat_edge_score<<<blocks, BT, 0, stream>>>(src, dst, s_src, s_dst, e, m, E);
    }
    // 5) exp + segment sum
    {
        int blocks = (E + BT - 1) / BT;
        gat_edge_exp<<<blocks, BT, 0, stream>>>(dst, m, e, denom, E);
    }
    // 6) weighted scatter-sum (one wave per edge)
    {
        int blocks = (E * 32 + BT - 1) / BT;
        gat_scatter<<<blocks, BT, 0, stream>>>(src, dst, e, denom, z, out, E);
    }
}